// nnChamferDis_77309411646
// MI455X (gfx1250) — compile-verified
//
#include <hip/hip_runtime.h>

// Chamfer distance via V_WMMA_F32_16X16X4_F32 on gfx1250 (wave32).
//
// d(m,n) = ||a_m||^2 + ( ||b_n||^2 - 2 a_m . b_n )
// The parenthesized term is produced as a 16x16 tile by ONE f32 K=4 WMMA
// using homogeneous coordinates:
//   A row m  = [-2ax, -2ay, -2az, 1]        (16x4)
//   B col n  = [ bx,   by,   bz, ||b||^2 ]  (4x16)
// Running row-min kept in 8 VGPRs (compiler fuses to v_min3_num_f32);
// ||a||^2 is min-invariant per row and added after the loop.

typedef __attribute__((ext_vector_type(2))) float v2f;
typedef __attribute__((ext_vector_type(8))) float v8f;

// NSTATIC > 0: compile-time point count (no unroll remainder / prologue).
// NSTATIC == 0: dynamic fallback.
template <int NSTATIC>
__global__ __launch_bounds__(256) void chamfer_min_kernel(
    const float* __restrict__ pc0, const float* __restrict__ pc1,
    float* __restrict__ ws, int Nrt)
{
    const int N = (NSTATIC > 0) ? NSTATIC : Nrt;

    // direction 0: queries=pc0 targets=pc1 ; direction 1: swapped
    const float* __restrict__ A = (blockIdx.y == 0) ? pc0 : pc1;
    const float* __restrict__ B = (blockIdx.y == 0) ? pc1 : pc0;

    const int lane  = threadIdx.x & 31;
    const int wave  = threadIdx.x >> 5;
    const int strip = blockIdx.x * 8 + wave;   // 16-row strip of the query cloud
    const int m0    = strip * 16;
    const bool hi   = (lane >= 16);            // lane half holds K=2/3 vs K=0/1
    const int sub   = lane & 15;

    // ---- A fragment (loaded once, reused for all column tiles) ----
    // 16x4 f32 A layout: lanes0-15 {V0=K0, V1=K1}, lanes16-31 {V0=K2, V1=K3}
    const float* ap = A + 3 * (m0 + sub);
    float ax = ap[0], ay = ap[1], az = ap[2];
    float na = ax * ax + ay * ay + az * az;    // ||a_m||^2 (folded in at the end)
    v2f amat;
    amat.x = hi ? (-2.0f * az) : (-2.0f * ax); // K2 : K0
    amat.y = hi ? 1.0f         : (-2.0f * ay); // K3 : K1

    v8f rmin;
#pragma unroll
    for (int k = 0; k < 8; ++k) rmin[k] = 3.0e38f;

    const float* bbase = B + 3 * sub;
    const v8f czero = {};

    // ---- stream all column tiles: 1 WMMA + 4 v_min3 per 256 pairs ----
#pragma unroll 8
    for (int n0 = 0; n0 < N; n0 += 16) {
        const float* bp = bbase + 3 * n0;      // folds to immediate offsets
        float bx = bp[0], by = bp[1], bz = bp[2];
        float nb = bx * bx + by * by + bz * bz;
        // 4x16 f32 B layout: lanes0-15 {V0=K0, V1=K1}, lanes16-31 {V0=K2, V1=K3}
        v2f bmat;
        bmat.x = hi ? bz : bx;                 // K2 : K0
        bmat.y = hi ? nb : by;                 // K3 : K1
        v8f d = __builtin_amdgcn_wmma_f32_16x16x4_f32(
            false, amat, false, bmat, (short)0, czero, false, false);
#pragma unroll
        for (int k = 0; k < 8; ++k) rmin[k] = fminf(rmin[k], d[k]);
    }

    // ---- finish: min across the 16 lanes of each half, add ||a||^2, sum rows
    // C/D layout: VGPR v, lanes0-15 -> row m0+v ; lanes16-31 -> row m0+v+8
    float s = 0.0f;
#pragma unroll
    for (int v = 0; v < 8; ++v) {
        float r = rmin[v];
        r = fminf(r, __shfl_xor(r, 1));
        r = fminf(r, __shfl_xor(r, 2));
        r = fminf(r, __shfl_xor(r, 4));
        r = fminf(r, __shfl_xor(r, 8));
        float nlo = __shfl(na, v);       // ||a_{m0+v}||^2
        float nhi = __shfl(na, v + 8);   // ||a_{m0+v+8}||^2
        s += r + (hi ? nhi : nlo);       // lanes0-15 sum rows 0..7, lanes16-31 rows 8..15
    }
    s += __shfl_xor(s, 16);              // total over the 16 rows of this strip
    if (lane == 0) ws[blockIdx.y * (gridDim.x * 8) + strip] = s;
}

// Deterministic fixed-tree reduction of the per-strip partial sums -> scalar.
__global__ __launch_bounds__(256) void chamfer_reduce_kernel(
    const float* __restrict__ ws, float* __restrict__ out, int count, float invN)
{
    __shared__ float sm[256];
    int t = threadIdx.x;
    float s = 0.0f;
    for (int i = t; i < count; i += 256) s += ws[i];
    sm[t] = s;
    __syncthreads();
    for (int w = 128; w > 0; w >>= 1) {
        if (t < w) sm[t] += sm[t + w];
        __syncthreads();
    }
    if (t == 0) out[0] = sm[0] * invN;   // mean0 + mean1, both clouds size N
}

extern "C" void kernel_launch(void* const* d_in, const int* in_sizes, int n_in,
                              void* d_out, int out_size, void* d_ws, size_t ws_size,
                              hipStream_t stream) {
    const float* pc0 = (const float*)d_in[0];
    const float* pc1 = (const float*)d_in[1];
    int N       = in_sizes[0] / 3;   // 32768 points per cloud (reference)
    int strips  = N / 16;            // row strips per direction
    int blocksX = strips / 8;        // 8 waves (strips) per 256-thread block
    float* ws   = (float*)d_ws;      // 2*strips floats of partial sums

    dim3 grid(blocksX, 2);           // y: direction (pc0->pc1, pc1->pc0)
    if (N == 32768) {
        chamfer_min_kernel<32768><<<grid, 256, 0, stream>>>(pc0, pc1, ws, N);
    } else {
        chamfer_min_kernel<0><<<grid, 256, 0, stream>>>(pc0, pc1, ws, N);
    }
    chamfer_reduce_kernel<<<1, 256, 0, stream>>>(ws, (float*)d_out,
                                                 2 * strips, 1.0f / (float)N);
}